// NeighborAggregation_17239998726507
// MI455X (gfx1250) — compile-verified
//
#include <hip/hip_runtime.h>
#include <cstddef>
#include <cstdint>

typedef float v2f __attribute__((ext_vector_type(2)));
typedef float v8f __attribute__((ext_vector_type(8)));

#define EMB   64
#define HID   64
#define KEY_M 64
#define NWAVE 13                 // 13 strips of 16 rows cover N=200 exactly (208 padded)
#define NTHR  (NWAVE*32)         // 416
#define S     68                 // padded LDS row stride in floats: 16B-aligned rows, bank stride 4
#define OFF_WQ  0
#define OFF_WK  (64*S)
#define OFF_WV  (2*64*S)
#define OFF_TK  (3*64*S)
#define OFF_SCR (4*64*S)         // union: key tile (64*S) before barrier2, per-wave tQ scratch (13*16*S) after
#define LDS_FLOATS (4*64*S + NWAVE*16*S)

__device__ __forceinline__ v8f wmma4(v2f a, v2f b, v8f c) {
  // D = A(16x4, f32) x B(4x16, f32) + C(16x16, f32)
  return __builtin_amdgcn_wmma_f32_16x16x4_f32(false, a, false, b, (short)0, c, false, false);
}

// Async copy 16B global -> LDS (CDNA5, tracked on ASYNCcnt). VDST = LDS byte offset.
__device__ __forceinline__ void async_ld128(const float* gsrc, float* ldst) {
  unsigned l = (unsigned)(uintptr_t)ldst;   // flat-LDS low 32 bits == wave-relative LDS offset
  asm volatile("global_load_async_to_lds_b128 %0, %1, off"
               :: "v"(l), "v"(gsrc) : "memory");
}

// Burst-load the 16 K-step fragments of one 16-wide LDS tile row-block into registers.
__device__ __forceinline__ void load_frags16(const float* lds, int rowbase, int kl, v2f* f) {
#pragma unroll
  for (int k = 0; k < 16; ++k)
    f[k] = *(const v2f*)&lds[rowbase + k * 4 + kl];
}

__global__ __launch_bounds__(NTHR, 1) void gated_attn_f32_wmma(
    const float* __restrict__ query,   // [B,N,64]
    const float* __restrict__ keyt,    // [B,64,64]
    const float* __restrict__ Qw,      // [64,64] row-major [h][e]
    const float* __restrict__ Kw,
    const float* __restrict__ Vw,
    float* __restrict__ out,           // [B,N,64]
    int N)
{
  extern __shared__ float lds[];
  const int tid  = threadIdx.x;
  const int lane = tid & 31;
  const int wave = tid >> 5;      // 0..12
  const int nlo  = lane & 15;     // M/N index within a 16-wide fragment
  const int hh   = lane >> 4;     // half-wave select (0/1)
  const int kl   = hh * 2;        // K sub-offset for f32 A/B fragments

  const int b = blockIdx.x;       // one workgroup per batch

  // ---- async-stage weights + key[b] tile straight into padded LDS (16B chunks) ----
  {
    const float* kb = keyt + (size_t)b * KEY_M * EMB;
    for (int idx = tid; idx < 64 * 16; idx += NTHR) {   // 1024 x 16B per matrix
      int e4 = idx & 15;                                // 16B column chunk
      int h  = idx >> 4;                                // row
      int ld = h * S + e4 * 4;                          // padded LDS float index (16B aligned)
      int gs = h * 64 + e4 * 4;                         // dense global float index
      async_ld128(Qw + gs, &lds[OFF_WQ  + ld]);
      async_ld128(Kw + gs, &lds[OFF_WK  + ld]);
      async_ld128(Vw + gs, &lds[OFF_WV  + ld]);
      async_ld128(kb + gs, &lds[OFF_SCR + ld]);
    }
  }
  asm volatile("s_wait_asynccnt 0x0" ::: "memory");
  __syncthreads();

  // ---- trans_K = key[b] @ Kw^T -> lds[OFF_TK][m][h]; 16 tiles of 16x16 over 13 waves ----
  for (int t = wave; t < 16; t += NWAVE) {
    int mi = t >> 2, hj = t & 3;
    v2f af[16], bf[16];
    load_frags16(lds, OFF_SCR + (mi*16 + nlo) * S, kl, af);   // A: key rows
    load_frags16(lds, OFF_WK  + (hj*16 + nlo) * S, kl, bf);   // B: Kw rows (B[e][h]=Kw[h][e])
    v8f acc = {};
#pragma unroll
    for (int k = 0; k < 16; ++k) acc = wmma4(af[k], bf[k], acc);
#pragma unroll
    for (int v = 0; v < 8; ++v)
      lds[OFF_TK + (mi*16 + v + hh*8)*S + hj*16 + nlo] = acc[v];
  }
  __syncthreads();

  // ---- per-wave 16-row strip of N ----
  const int n0 = wave * 16;

  // A fragments from query (row-clamped for the N=200 tail; invalid rows never stored)
  int qrow = n0 + nlo; if (qrow >= N) qrow = N - 1;
  const float* qbase = query + ((size_t)b * N + qrow) * EMB;
  v2f a[16];
#pragma unroll
  for (int k = 0; k < 16; ++k)
    a[k] = *(const v2f*)(qbase + k*4 + kl);

  // tV and tQ projections: burst-load both B tiles, then two interleaved WMMA chains
  v8f accV[4], accQ[4];
#pragma unroll
  for (int t = 0; t < 4; ++t) { accV[t] = (v8f){}; accQ[t] = (v8f){}; }
#pragma unroll
  for (int t = 0; t < 4; ++t) {
    v2f bq[16], bv[16];
    load_frags16(lds, OFF_WQ + (t*16 + nlo) * S, kl, bq);
    load_frags16(lds, OFF_WV + (t*16 + nlo) * S, kl, bv);
#pragma unroll
    for (int k = 0; k < 16; ++k) {
      accQ[t] = wmma4(a[k], bq[k], accQ[t]);
      accV[t] = wmma4(a[k], bv[k], accV[t]);
    }
  }

  // C-layout -> A-layout round trip for tQ through per-wave LDS scratch
  const int scr = OFF_SCR + wave * 16 * S;
#pragma unroll
  for (int t = 0; t < 4; ++t)
#pragma unroll
    for (int v = 0; v < 8; ++v)
      lds[scr + (v + hh*8)*S + t*16 + nlo] = accQ[t][v];
  __asm__ volatile("s_wait_dscnt 0x0" ::: "memory");  // same-wave LDS RAW fence

  v2f a2[16];
  load_frags16(lds, scr + nlo * S, kl, a2);

  // logits = tQ @ tK^T  (B[h][m] = tK[m][h], read from [m][h] rows)
  v8f accL[4];
#pragma unroll
  for (int t = 0; t < 4; ++t) accL[t] = (v8f){};
#pragma unroll
  for (int t = 0; t < 4; ++t) {
    v2f bt[16];
    load_frags16(lds, OFF_TK + (t*16 + nlo) * S, kl, bt);
#pragma unroll
    for (int k = 0; k < 16; ++k)
      accL[t] = wmma4(a2[k], bt[k], accL[t]);
  }

  // ---- softmax over M=64 (row lives in 16 lanes of one half-wave) + gate + store ----
  const float scale = 0.125f;  // 1/sqrt(64)
#pragma unroll
  for (int v = 0; v < 8; ++v) {
    float m = accL[0][v];
#pragma unroll
    for (int t = 1; t < 4; ++t) m = fmaxf(m, accL[t][v]);
    m *= scale;
#pragma unroll
    for (int off = 1; off < 16; off <<= 1)
      m = fmaxf(m, __shfl_xor(m, off, 32));
    float s = 0.0f;
#pragma unroll
    for (int t = 0; t < 4; ++t) {
      float e = __expf(accL[t][v] * scale - m);
      accL[t][v] = e;
      s += e;
    }
#pragma unroll
    for (int off = 1; off < 16; off <<= 1)
      s += __shfl_xor(s, off, 32);
    float inv = __frcp_rn(s);

    int n = n0 + v + hh * 8;
    if (n < N) {
      float* orow = out + ((size_t)b * N + n) * HID;
#pragma unroll
      for (int t = 0; t < 4; ++t)
        orow[t*16 + nlo] = accV[t][v] * (accL[t][v] * inv);
    }
  }
}

extern "C" void kernel_launch(void* const* d_in, const int* in_sizes, int n_in,
                              void* d_out, int out_size, void* d_ws, size_t ws_size,
                              hipStream_t stream) {
  const float* query = (const float*)d_in[0];   // [B,200,64]
  const float* keyt  = (const float*)d_in[1];   // [B,64,64]
  const float* Qw    = (const float*)d_in[2];   // [64,64]
  const float* Kw    = (const float*)d_in[3];
  const float* Vw    = (const float*)d_in[4];
  float* out = (float*)d_out;

  const int N = 200;
  const int B = in_sizes[0] / (N * EMB);

  dim3 grid(B);         // one workgroup per batch
  dim3 block(NTHR);     // 13 wave32's, one 16-row strip each
  size_t ldsBytes = (size_t)LDS_FLOATS * sizeof(float);
  hipLaunchKernelGGL(gated_attn_f32_wmma, grid, block, ldsBytes, stream,
                     query, keyt, Qw, Kw, Vw, out, N);
}